// Loss_14139032338660
// MI455X (gfx1250) — compile-verified
//
#include <hip/hip_runtime.h>

typedef __attribute__((ext_vector_type(2))) float v2f;
typedef __attribute__((ext_vector_type(4))) float v4f;
typedef __attribute__((ext_vector_type(8))) float v8f;
typedef __attribute__((ext_vector_type(4))) int   v4i;

#define NPARTS 2048              // grid size; multiple of 64 for WMMA combine
#define LOG_CLAMP_F (-100.0f)

// Stage 1: streaming pass. Each thread consumes float4/int4 chunks with
// non-temporal b128 loads, computes one v_log_f32 per element, and keeps
// three fp32 partials (sum of clamped log over positives, over negatives,
// and the positive count). Per-block tree reduction -> one partial triple
// per block in the workspace.
__global__ __launch_bounds__(256) void bce_partials_kernel(
    const float* __restrict__ x, const int* __restrict__ gt,
    float* __restrict__ pA, float* __restrict__ pB, float* __restrict__ pS,
    long long n)
{
    const long long nvec   = n >> 2;
    const long long stride = (long long)gridDim.x * blockDim.x;
    long long i = (long long)blockIdx.x * blockDim.x + threadIdx.x;

    const v4f* __restrict__ x4 = (const v4f*)x;
    const v4i* __restrict__ g4 = (const v4i*)gt;

    float accA = 0.0f, accB = 0.0f, accS = 0.0f;

    for (; i < nvec; i += stride) {
        v4f xv = __builtin_nontemporal_load(x4 + i);
        v4i gv = __builtin_nontemporal_load(g4 + i);
        // speculative prefetch of next grid-stride chunk (dropped if OOB)
        __builtin_prefetch((const void*)(x4 + i + stride), 0, 0);
        __builtin_prefetch((const void*)(g4 + i + stride), 0, 0);
#pragma unroll
        for (int j = 0; j < 4; ++j) {
            float xj  = xv[j];
            bool  p   = gv[j] != 0;
            float arg = p ? xj : (1.0f - xj);            // only one log needed
            float l   = fmaxf(__logf(arg), LOG_CLAMP_F); // v_log_f32; log(0)=-inf -> -100
            accA += p ? l    : 0.0f;
            accB += p ? 0.0f : l;
            accS += p ? 1.0f : 0.0f;                     // exact: counts < 2^24
        }
    }

    // scalar tail (N % 4), handled by block 0
    if (blockIdx.x == 0) {
        for (long long t = (nvec << 2) + threadIdx.x; t < n; t += blockDim.x) {
            float xj  = x[t];
            bool  p   = gt[t] != 0;
            float arg = p ? xj : (1.0f - xj);
            float l   = fmaxf(__logf(arg), LOG_CLAMP_F);
            accA += p ? l    : 0.0f;
            accB += p ? 0.0f : l;
            accS += p ? 1.0f : 0.0f;
        }
    }

    // wave32 tree reduction
#pragma unroll
    for (int off = 16; off > 0; off >>= 1) {
        accA += __shfl_down(accA, off, 32);
        accB += __shfl_down(accB, off, 32);
        accS += __shfl_down(accS, off, 32);
    }

    __shared__ float sA[8], sB[8], sS[8];
    const int wave = threadIdx.x >> 5;
    const int lane = threadIdx.x & 31;
    if (lane == 0) { sA[wave] = accA; sB[wave] = accB; sS[wave] = accS; }
    __syncthreads();
    if (threadIdx.x == 0) {
        float a = 0.0f, b = 0.0f, s = 0.0f;
#pragma unroll
        for (int w = 0; w < 8; ++w) { a += sA[w]; b += sB[w]; s += sS[w]; }
        pA[blockIdx.x] = a; pB[blockIdx.x] = b; pS[blockIdx.x] = s;
    }
}

// Stage 2: single wave. Sum NPARTS partials per array with
// V_WMMA_F32_16X16X4_F32 using an all-ones B: D[m][n] = rowsum(A[m][:]) + C,
// i.e. a fused 64-way f32 reduction per issue, fully deterministic.
// A 16x4 layout: lanes 0-15 hold (M=lane, K=0..1), lanes 16-31 (M=lane-16, K=2..3).
// We map A[m][k] = p[base + m*4 + k] (any bijection works for a total sum).
// D layout: lane n<16 VGPR j = D[j][n]; lane n>=16 VGPR j = D[8+j][n-16], so
// total = (sum of 8 regs in lane 0) + (sum of 8 regs in lane 16).
__global__ __launch_bounds__(32) void bce_combine_kernel(
    const float* __restrict__ pA, const float* __restrict__ pB,
    const float* __restrict__ pS, float* __restrict__ out, float nf)
{
    const int lane = threadIdx.x;
    const int row  = lane & 15;
    const int kb   = (lane >> 4) << 1;   // 0 for lanes 0-15, 2 for lanes 16-31

    v2f ones; ones[0] = 1.0f; ones[1] = 1.0f;
    v8f cA = {}; v8f cB = {}; v8f cS = {};

    for (int base = 0; base < NPARTS; base += 64) {
        const int i0 = base + row * 4 + kb;
        v2f a, b, s;
        a[0] = pA[i0]; a[1] = pA[i0 + 1];
        b[0] = pB[i0]; b[1] = pB[i0 + 1];
        s[0] = pS[i0]; s[1] = pS[i0 + 1];
        cA = __builtin_amdgcn_wmma_f32_16x16x4_f32(false, a, false, ones, (short)0, cA, false, false);
        cB = __builtin_amdgcn_wmma_f32_16x16x4_f32(false, b, false, ones, (short)0, cB, false, false);
        cS = __builtin_amdgcn_wmma_f32_16x16x4_f32(false, s, false, ones, (short)0, cS, false, false);
    }

    float tA = 0.0f, tB = 0.0f, tS = 0.0f;
#pragma unroll
    for (int j = 0; j < 8; ++j) { tA += cA[j]; tB += cB[j]; tS += cS[j]; }

    const float sumA = __shfl(tA, 0, 32) + __shfl(tA, 16, 32);
    const float sumB = __shfl(tB, 0, 32) + __shfl(tB, 16, 32);
    const float sumS = __shfl(tS, 0, 32) + __shfl(tS, 16, 32);

    if (lane == 0) {
        const float s = sumS;
        // loss = -A/(2s) - B/(2(n-s))
        out[0] = -sumA / (2.0f * s) - sumB / (2.0f * (nf - s));
    }
}

extern "C" void kernel_launch(void* const* d_in, const int* in_sizes, int n_in,
                              void* d_out, int out_size, void* d_ws, size_t ws_size,
                              hipStream_t stream)
{
    const float* x  = (const float*)d_in[0];
    const int*   gt = (const int*)d_in[1];
    const long long n = (long long)in_sizes[0];

    float* pA = (float*)d_ws;          // 3 * NPARTS * 4 = 24 KB of workspace
    float* pB = pA + NPARTS;
    float* pS = pB + NPARTS;

    bce_partials_kernel<<<NPARTS, 256, 0, stream>>>(x, gt, pA, pB, pS, n);
    bce_combine_kernel<<<1, 32, 0, stream>>>(pA, pB, pS, (float*)d_out, (float)n);
}